// MomGatedLinearAttention_54434415510220
// MI455X (gfx1250) — compile-verified
//
#include <hip/hip_runtime.h>
#include <math.h>

// ---------------- model constants ----------------
#define Bc 4
#define Sc 2048
#define Dc 1024
#define Mc 4
#define CAPc 1024
#define DKc 512
#define DVc 1024
#define Hc 4
#define HDKc 128
#define HDVc 256
#define LOWc 16
#define GATE_NORM_INV 0.0625f
#define EPSc 1e-5f

typedef float v2f __attribute__((ext_vector_type(2)));
typedef float v8f __attribute__((ext_vector_type(8)));
typedef int   v4i __attribute__((__vector_size__(16)));

#define AS1 __attribute__((address_space(1)))
#define AS3 __attribute__((address_space(3)))

#if __has_builtin(__builtin_amdgcn_global_load_async_to_lds_b128) && \
    __has_builtin(__builtin_amdgcn_s_wait_asynccnt)
#define HAS_ASYNC_LDS 1
#else
#define HAS_ASYNC_LDS 0
#endif

// =====================================================================
// 1) Router: softmax over 4 memories + top-2, per-batch memory counts
// =====================================================================
__global__ __launch_bounds__(256) void k_router(const float* __restrict__ x,
                                                const float* __restrict__ wg,
                                                float* __restrict__ rw,
                                                int* __restrict__ sel,
                                                int* __restrict__ counts)
{
    int tid  = threadIdx.x;
    int lane = tid & 31;
    int wid  = tid >> 5;
    int tok  = blockIdx.x * 8 + wid;          // < B*S
    const float* xr = x + (size_t)tok * Dc;
    float a0 = 0.f, a1 = 0.f, a2 = 0.f, a3 = 0.f;
    for (int d = lane; d < Dc; d += 32) {
        float xv = xr[d];
        const float* w = wg + (size_t)d * Mc;
        a0 = fmaf(xv, w[0], a0);
        a1 = fmaf(xv, w[1], a1);
        a2 = fmaf(xv, w[2], a2);
        a3 = fmaf(xv, w[3], a3);
    }
    for (int off = 16; off > 0; off >>= 1) {
        a0 += __shfl_xor(a0, off, 32);
        a1 += __shfl_xor(a1, off, 32);
        a2 += __shfl_xor(a2, off, 32);
        a3 += __shfl_xor(a3, off, 32);
    }
    if (lane == 0) {
        float l[4] = {a0, a1, a2, a3};
        float mx = fmaxf(fmaxf(l[0], l[1]), fmaxf(l[2], l[3]));
        float p[4]; float s = 0.f;
        for (int i = 0; i < 4; i++) { p[i] = expf(l[i] - mx); s += p[i]; }
        float inv = 1.f / s;
        for (int i = 0; i < 4; i++) p[i] *= inv;
        int i0 = 0;
        for (int i = 1; i < 4; i++) if (p[i] > p[i0]) i0 = i;   // first-max tie rule
        int i1 = (i0 == 0) ? 1 : 0;
        for (int i = 0; i < 4; i++) if (i != i0 && p[i] > p[i1]) i1 = i;
        float w0 = p[i0], w1 = p[i1], wsum = w0 + w1;
        sel[tok * 2 + 0] = i0;
        sel[tok * 2 + 1] = i1;
        rw[tok * 2 + 0] = w0 / wsum;
        rw[tok * 2 + 1] = w1 / wsum;
        int b = tok / Sc;
        atomicAdd(&counts[b * Mc + i0], 1);
        atomicAdd(&counts[b * Mc + i1], 1);
    }
}

// =====================================================================
// 2) init: tok map = -1, counts = 0  (deterministic every launch)
// =====================================================================
__global__ __launch_bounds__(256) void k_init(int* __restrict__ tok, int* __restrict__ counts)
{
    int i = blockIdx.x * 256 + threadIdx.x;
    if (i < Mc * Bc * CAPc) tok[i] = -1;
    if (i < Bc * Mc)        counts[i] = 0;
}

// =====================================================================
// 3) Stable rank scan per (b, mem); capacity-truncated slot assignment
//    slot j = rank + CAP - count   (keep most recent CAP tokens)
// =====================================================================
__global__ __launch_bounds__(32) void k_scan(const int* __restrict__ sel,
                                             const int* __restrict__ counts,
                                             int* __restrict__ slot,
                                             int* __restrict__ tok)
{
    int b = blockIdx.x / Mc;
    int m = blockIdx.x % Mc;
    int lane = threadIdx.x;
    int c = counts[b * Mc + m];
    int base = 0;
    for (int t0 = 0; t0 < Sc; t0 += 32) {
        int t  = t0 + lane;
        int s0 = sel[(size_t)(b * Sc + t) * 2 + 0];
        int s1 = sel[(size_t)(b * Sc + t) * 2 + 1];
        int kk = (s0 == m) ? 0 : ((s1 == m) ? 1 : -1);
        int ind = (kk >= 0) ? 1 : 0;
        int v = ind;
        for (int off = 1; off < 32; off <<= 1) {
            int u = __shfl_up(v, off, 32);
            if (lane >= off) v += u;
        }
        int total = __shfl(v, 31, 32);
        if (kk >= 0) {
            int p = base + v - 1;                  // exclusive rank within (b,m)
            int j = p + CAPc - c;                  // capacity left-truncation
            if (j >= 0) {
                slot[(size_t)(b * Sc + t) * 2 + kk] = j;
                tok[(size_t)(m * Bc + b) * CAPc + j] = t;
            } else {
                slot[(size_t)(b * Sc + t) * 2 + kk] = -1;
            }
        }
        base += total;
    }
}

// =====================================================================
// 4) Gather dispatched tokens into tx (M,b,cap,D), zero padded slots
// =====================================================================
__global__ __launch_bounds__(256) void k_gather(const float* __restrict__ x,
                                                const int* __restrict__ tok,
                                                float* __restrict__ tx)
{
    int row = blockIdx.x;                         // (m*B + b)*CAP + j
    int t = tok[row];
    int b = (row / CAPc) % Bc;
    float4 v = {0.f, 0.f, 0.f, 0.f};
    if (t >= 0)
        v = *(const float4*)(x + ((size_t)(b * Sc + t)) * Dc + threadIdx.x * 4);
    *(float4*)(tx + (size_t)row * Dc + threadIdx.x * 4) = v;
}

// =====================================================================
// 5) f32 WMMA GEMM  C[R,N] = A[R,K] * W[K,N]  (+ fused epilogue)
//    Block tile 64x128, wave tile 32x32 (4 accumulators), KT=32,
//    double-buffered LDS staged with GLOBAL_LOAD_ASYNC_TO_LDS_B128
//    (ASYNCcnt) when available.
//    act 0: val * scale      act 1: logsigmoid(val + bias[col]) / 16
// =====================================================================
#define BM 64
#define BN 128
#define KTl 32
#define ALD (KTl + 4)   // 36 floats: 144B row stride (16B multiple, conflict-free)
#define WLD (BN + 4)    // 132 floats: 528B row stride (16B multiple)

__device__ __forceinline__ float gemm_epilogue(float v, int col,
                                               const float* bias, float scale, int act)
{
    if (act == 1) {
        float z  = v + bias[col];
        float ls = fminf(z, 0.f) - log1pf(__expf(-fabsf(z)));   // stable logsigmoid
        return ls * GATE_NORM_INV;
    }
    return v * scale;
}

__global__ __launch_bounds__(256) void k_gemm(const float* __restrict__ A,
                                              const float* __restrict__ W,
                                              float* __restrict__ C,
                                              const float* __restrict__ bias,
                                              int R, int K, int N,
                                              float scale, int act)
{
    __shared__ __attribute__((aligned(16))) float As[2][BM][ALD];
    __shared__ __attribute__((aligned(16))) float Ws[2][KTl][WLD];

    int tid  = threadIdx.x;
    int lane = tid & 31;
    int wid  = tid >> 5;
    int wr   = wid & 1;            // wave row strip (32 rows)
    int wc   = wid >> 1;           // wave col strip (32 cols), 0..3
    int row0 = blockIdx.y * BM;
    int col0 = blockIdx.x * BN;
    int half = lane >> 4;          // K split: lanes 0-15 -> K{0,1}, 16-31 -> K{2,3}
    int l15  = lane & 15;

    v8f acc00 = {0.f,0.f,0.f,0.f,0.f,0.f,0.f,0.f};
    v8f acc01 = {0.f,0.f,0.f,0.f,0.f,0.f,0.f,0.f};
    v8f acc10 = {0.f,0.f,0.f,0.f,0.f,0.f,0.f,0.f};
    v8f acc11 = {0.f,0.f,0.f,0.f,0.f,0.f,0.f,0.f};

    // staging thread maps
    int ar  = tid >> 2;            // A row 0..63
    int ac0 = (tid & 3) << 3;      // A k base 0,8,16,24 (2 x float4 each)
    int wkr = tid >> 5;            // W k row 0..7 (4 rows per thread, stride 8)
    int wn  = (tid & 31) << 2;     // W col 0..124

    auto stage = [&](int buf, int k0) {
        #pragma unroll
        for (int e = 0; e < 2; ++e) {
            int kc = ac0 + e * 4;
            float* dst = &As[buf][ar][kc];
            if (k0 + kc + 3 < K) {
                const float* src = A + (size_t)(row0 + ar) * K + k0 + kc;
#if HAS_ASYNC_LDS
                __builtin_amdgcn_global_load_async_to_lds_b128(
                    (AS1 v4i*)src, (AS3 v4i*)dst, 0, 0);
#else
                float4 v = *(const float4*)src;
                dst[0] = v.x; dst[1] = v.y; dst[2] = v.z; dst[3] = v.w;
#endif
            } else {
                dst[0] = 0.f; dst[1] = 0.f; dst[2] = 0.f; dst[3] = 0.f;
            }
        }
        #pragma unroll
        for (int e = 0; e < 4; ++e) {
            int kr = wkr + e * 8;
            float* dst = &Ws[buf][kr][wn];
            if ((k0 + kr < K) && (col0 + wn + 3 < N)) {
                const float* src = W + (size_t)(k0 + kr) * N + col0 + wn;
#if HAS_ASYNC_LDS
                __builtin_amdgcn_global_load_async_to_lds_b128(
                    (AS1 v4i*)src, (AS3 v4i*)dst, 0, 0);
#else
                float4 v = *(const float4*)src;
                dst[0] = v.x; dst[1] = v.y; dst[2] = v.z; dst[3] = v.w;
#endif
            } else {
                float tv[4];
                #pragma unroll
                for (int q2 = 0; q2 < 4; ++q2)
                    tv[q2] = ((k0 + kr < K) && (col0 + wn + q2 < N))
                           ? W[(size_t)(k0 + kr) * N + col0 + wn + q2] : 0.f;
                dst[0] = tv[0]; dst[1] = tv[1]; dst[2] = tv[2]; dst[3] = tv[3];
            }
        }
    };

    stage(0, 0);
#if HAS_ASYNC_LDS
    __builtin_amdgcn_s_wait_asynccnt(0);
#endif
    __syncthreads();

    int nstage = (K + KTl - 1) / KTl;
    for (int s = 0; s < nstage; ++s) {
        int buf = s & 1;
        if (s + 1 < nstage) stage(buf ^ 1, (s + 1) * KTl);   // overlap with compute
        #pragma unroll
        for (int kk = 0; kk < KTl; kk += 4) {
            int ksel = kk + 2 * half;
            v2f a0v, a1v, b0v, b1v;
            a0v.x = As[buf][wr * 32 + l15][ksel + 0];
            a0v.y = As[buf][wr * 32 + l15][ksel + 1];
            a1v.x = As[buf][wr * 32 + 16 + l15][ksel + 0];
            a1v.y = As[buf][wr * 32 + 16 + l15][ksel + 1];
            b0v.x = Ws[buf][ksel + 0][wc * 32 + l15];
            b0v.y = Ws[buf][ksel + 1][wc * 32 + l15];
            b1v.x = Ws[buf][ksel + 0][wc * 32 + 16 + l15];
            b1v.y = Ws[buf][ksel + 1][wc * 32 + 16 + l15];
            acc00 = __builtin_amdgcn_wmma_f32_16x16x4_f32(false, a0v, false, b0v,
                                                          (short)0, acc00, false, false);
            acc01 = __builtin_amdgcn_wmma_f32_16x16x4_f32(false, a0v, false, b1v,
                                                          (short)0, acc01, false, false);
            acc10 = __builtin_amdgcn_wmma_f32_16x16x4_f32(false, a1v, false, b0v,
                                                          (short)0, acc10, false, false);
            acc11 = __builtin_amdgcn_wmma_f32_16x16x4_f32(false, a1v, false, b1v,
                                                          (short)0, acc11, false, false);
        }
#if HAS_ASYNC_LDS
        __builtin_amdgcn_s_wait_asynccnt(0);
#endif
        __syncthreads();
    }

    int colA = col0 + wc * 32 + l15;
    int colB = colA + 16;
    int rb0  = row0 + wr * 32 + half * 8;        // rows for acc0x
    int rb1  = rb0 + 16;                         // rows for acc1x
    #pragma unroll
    for (int r = 0; r < 8; r++) {
        if (colA < N) {
            C[(size_t)(rb0 + r) * N + colA] = gemm_epilogue(acc00[r], colA, bias, scale, act);
            C[(size_t)(rb1 + r) * N + colA] = gemm_epilogue(acc10[r], colA, bias, scale, act);
        }
        if (colB < N) {
            C[(size_t)(rb0 + r) * N + colB] = gemm_epilogue(acc01[r], colB, bias, scale, act);
            C[(size_t)(rb1 + r) * N + colB] = gemm_epilogue(acc11[r], colB, bias, scale, act);
        }
    }
    (void)R;
}

// =====================================================================
// 6) Gated linear attention scan. One block per (state, head):
//    state S[128x256] in VGPRs: lane owns one v-column, 128 regs.
//    q/k/exp(gk) broadcast through double-buffered LDS.
// =====================================================================
__global__ __launch_bounds__(256) void k_gla(const float* __restrict__ q,
                                             const float* __restrict__ k,
                                             const float* __restrict__ v,
                                             const float* __restrict__ gk,
                                             float* __restrict__ y,
                                             int T)
{
    __shared__ float sq[2][HDKc];
    __shared__ float sk[2][HDKc];
    __shared__ float se[2][HDKc];

    int h = blockIdx.x;
    size_t rowbase = (size_t)blockIdx.y * T;
    int tid = threadIdx.x;

    float S[HDKc];
    #pragma unroll
    for (int i = 0; i < HDKc; i++) S[i] = 0.f;

    for (int t = 0; t < T; ++t) {
        int buf = t & 1;
        size_t rk = (rowbase + t) * DKc + (size_t)h * HDKc;
        size_t rv = (rowbase + t) * DVc + (size_t)h * HDVc;
        if (tid < HDKc) {
            sq[buf][tid] = q[rk + tid];
            se[buf][tid] = __expf(gk[rk + tid]);
        } else {
            sk[buf][tid - HDKc] = k[rk + (tid - HDKc)];
        }
        float vj = v[rv + tid];
        if (t + 1 < T) {
            if (tid < HDKc) {
                __builtin_prefetch(q  + rk + DKc + tid, 0, 0);
                __builtin_prefetch(gk + rk + DKc + tid, 0, 0);
            } else {
                __builtin_prefetch(k + rk + DKc + (tid - HDKc), 0, 0);
            }
            __builtin_prefetch(v + rv + DVc + tid, 0, 0);
        }
        __syncthreads();
        float acc = 0.f;
        #pragma unroll
        for (int i = 0; i < HDKc; i++) {
            float s = fmaf(S[i], se[buf][i], sk[buf][i] * vj);
            S[i] = s;
            acc = fmaf(sq[buf][i], s, acc);
        }
        y[rv + tid] = acc;
        // double-buffered LDS: next iteration writes buf^1, safe with one barrier
    }
}

// =====================================================================
// 7) Recombine routed + shared, per-head RMSNorm, swish gate
// =====================================================================
__global__ __launch_bounds__(256) void k_combine(const float* __restrict__ ys,
                                                 const float* __restrict__ yr,
                                                 const float* __restrict__ g,
                                                 const float* __restrict__ rw,
                                                 const int* __restrict__ sel,
                                                 const int* __restrict__ slot,
                                                 const float* __restrict__ norm_w,
                                                 float* __restrict__ opre)
{
    __shared__ float red[256];
    int tok = blockIdx.x;            // b*S + t
    int b   = tok / Sc;
    int tid = threadIdx.x;
    size_t o4 = (size_t)tok * DVc + tid * 4;

    float4 o = *(const float4*)(ys + o4);
    #pragma unroll
    for (int kk = 0; kk < 2; ++kk) {
        int m = sel[tok * 2 + kk];
        int j = slot[tok * 2 + kk];
        float w = rw[tok * 2 + kk];
        if (j >= 0) {
            float4 yv = *(const float4*)(yr + ((size_t)(m * Bc + b) * CAPc + j) * DVc + tid * 4);
            o.x = fmaf(w, yv.x, o.x);
            o.y = fmaf(w, yv.y, o.y);
            o.z = fmaf(w, yv.z, o.z);
            o.w = fmaf(w, yv.w, o.w);
        }
    }
    float ss = o.x * o.x + o.y * o.y + o.z * o.z + o.w * o.w;
    red[tid] = ss;
    __syncthreads();
    for (int off = 32; off > 0; off >>= 1) {
        if ((tid & 63) < off) red[tid] += red[tid + off];
        __syncthreads();
    }
    float rms = rsqrtf(red[tid & ~63] * (1.0f / HDVc) + EPSc);

    int dl = (tid & 63) * 4;
    float4 nw = *(const float4*)(norm_w + dl);
    float4 gv = *(const float4*)(g + o4);
    o.x *= rms * nw.x; o.y *= rms * nw.y; o.z *= rms * nw.z; o.w *= rms * nw.w;
    o.x *= gv.x / (1.f + __expf(-gv.x));
    o.y *= gv.y / (1.f + __expf(-gv.y));
    o.z *= gv.z / (1.f + __expf(-gv.z));
    o.w *= gv.w / (1.f + __expf(-gv.w));
    *(float4*)(opre + o4) = o;
}

// =====================================================================
// launch
// =====================================================================
extern "C" void kernel_launch(void* const* d_in, const int* in_sizes, int n_in,
                              void* d_out, int out_size, void* d_ws, size_t ws_size,
                              hipStream_t stream)
{
    const float* x      = (const float*)d_in[0];
    const float* w_gate = (const float*)d_in[1];
    const float* w_q    = (const float*)d_in[2];
    const float* w_k    = (const float*)d_in[3];
    const float* w_v    = (const float*)d_in[4];
    const float* w_gk1  = (const float*)d_in[5];
    const float* w_gk2  = (const float*)d_in[6];
    const float* b_gk   = (const float*)d_in[7];
    const float* w_ks   = (const float*)d_in[8];
    const float* w_vs   = (const float*)d_in[9];
    const float* w_gk1s = (const float*)d_in[10];
    const float* w_gk2s = (const float*)d_in[11];
    const float* b_gks  = (const float*)d_in[12];
    const float* w_g    = (const float*)d_in[13];
    const float* w_o    = (const float*)d_in[14];
    const float* norm_w = (const float*)d_in[15];
    float* out = (float*)d_out;

    char* ws = (char*)d_ws;
    size_t off = 0;
    auto carve = [&](size_t bytes) -> void* {
        void* p = ws + off;
        off += (bytes + 255) & ~(size_t)255;
        return p;
    };
    const int RT = Mc * Bc * CAPc;   // 16384 routed rows
    const int RS = Bc * Sc;          // 8192 shared rows

    float* d_rw   = (float*)carve((size_t)RS * 2 * sizeof(float));
    int*   d_sel  = (int*)  carve((size_t)RS * 2 * sizeof(int));
    int*   d_slot = (int*)  carve((size_t)RS * 2 * sizeof(int));
    int*   d_cnt  = (int*)  carve((size_t)Bc * Mc * sizeof(int));
    int*   d_tok  = (int*)  carve((size_t)RT * sizeof(int));
    float* d_tx   = (float*)carve((size_t)RT * Dc * sizeof(float));
    float* d_qr   = (float*)carve((size_t)RT * DKc * sizeof(float));
    float* d_kr   = (float*)carve((size_t)RT * DKc * sizeof(float));
    float* d_vr   = (float*)carve((size_t)RT * DVc * sizeof(float));
    float* d_gklr = (float*)carve((size_t)RT * LOWc * sizeof(float));
    float* d_gkr  = (float*)carve((size_t)RT * DKc * sizeof(float));
    float* d_yr   = (float*)carve((size_t)RT * DVc * sizeof(float));
    float* d_qs   = (float*)carve((size_t)RS * DKc * sizeof(float));
    float* d_ks   = (float*)carve((size_t)RS * DKc * sizeof(float));
    float* d_vs   = (float*)carve((size_t)RS * DVc * sizeof(float));
    float* d_gkls = (float*)carve((size_t)RS * LOWc * sizeof(float));
    float* d_gks  = (float*)carve((size_t)RS * DKc * sizeof(float));
    float* d_ys   = (float*)carve((size_t)RS * DVc * sizeof(float));
    // tx is dead after routed projections: alias gate proj + normalized output
    float* d_g    = d_tx;
    float* d_opre = d_tx + (size_t)RS * DVc;

    const float qscale = 0.08838834764831843f;   // 128^-0.5
    dim3 blk(256);

    k_init  <<<dim3((RT + 255) / 256), blk, 0, stream>>>(d_tok, d_cnt);
    k_router<<<dim3(RS / 8),           blk, 0, stream>>>(x, w_gate, d_rw, d_sel, d_cnt);
    k_scan  <<<dim3(Bc * Mc), dim3(32), 0, stream>>>(d_sel, d_cnt, d_slot, d_tok);
    k_gather<<<dim3(RT),               blk, 0, stream>>>(x, d_tok, d_tx);

    auto gemm = [&](const float* A, const float* W2, float* C, const float* bias,
                    int R, int K, int N, float scale, int act) {
        dim3 g((N + BN - 1) / BN, R / BM);
        k_gemm<<<g, blk, 0, stream>>>(A, W2, C, bias, R, K, N, scale, act);
    };

    // routed projections (per-memory weights; q shared)
    gemm(d_tx, w_q, d_qr, nullptr, RT, Dc, DKc, qscale, 0);
    for (int m = 0; m < Mc; m++) {
        const float* Am = d_tx + (size_t)m * Bc * CAPc * Dc;
        int Rm = Bc * CAPc;
        gemm(Am, w_k   + (size_t)m * Dc * DKc,  d_kr   + (size_t)m * Rm * DKc,  nullptr, Rm, Dc,   DKc,  1.f, 0);
        gemm(Am, w_v   + (size_t)m * Dc * DVc,  d_vr   + (size_t)m * Rm * DVc,  nullptr, Rm, Dc,   DVc,  1.f, 0);
        gemm(Am, w_gk1 + (size_t)m * Dc * LOWc, d_gklr + (size_t)m * Rm * LOWc, nullptr, Rm, Dc,   LOWc, 1.f, 0);
        gemm(d_gklr + (size_t)m * Rm * LOWc, w_gk2 + (size_t)m * LOWc * DKc,
             d_gkr + (size_t)m * Rm * DKc, b_gk + (size_t)m * DKc, Rm, LOWc, DKc, 1.f, 1);
    }
    // shared (always-on) projections
    gemm(x, w_q,    d_qs,   nullptr, RS, Dc,   DKc,  qscale, 0);
    gemm(x, w_ks,   d_ks,   nullptr, RS, Dc,   DKc,  1.f,    0);
    gemm(x, w_vs,   d_vs,   nullptr, RS, Dc,   DVc,  1.f,    0);
    gemm(x, w_gk1s, d_gkls, nullptr, RS, Dc,   LOWc, 1.f,    0);
    gemm(d_gkls, w_gk2s, d_gks, b_gks, RS, LOWc, DKc, 1.f,   1);

    // gated linear attention scans
    k_gla<<<dim3(Hc, Mc * Bc), blk, 0, stream>>>(d_qr, d_kr, d_vr, d_gkr, d_yr, CAPc);
    k_gla<<<dim3(Hc, Bc),      blk, 0, stream>>>(d_qs, d_ks, d_vs, d_gks, d_ys, Sc);

    // gate projection (overwrites dead tx region), recombine + norm + gate
    gemm(x, w_g, d_g, nullptr, RS, Dc, DVc, 1.f, 0);
    k_combine<<<dim3(RS), blk, 0, stream>>>(d_ys, d_yr, d_g, d_rw, d_sel, d_slot, norm_w, d_opre);

    // final output projection
    gemm(d_opre, w_o, out, nullptr, RS, DVc, Dc, 1.f, 0);

    (void)in_sizes; (void)n_in; (void)out_size; (void)ws_size;
}